// ScoreBlock_7662221656315
// MI455X (gfx1250) — compile-verified
//
#include <hip/hip_runtime.h>

typedef float v2f __attribute__((ext_vector_type(2)));
typedef float v8f __attribute__((ext_vector_type(8)));

#define NSTOCK 500
#define FEAT   1502   // 2 + 3*500
#define WAVES  8
#define TPB    (WAVES * 32)

// Branchless exact-GELU: 0.5*x*(1+erf(x/sqrt(2))) with erf via
// Abramowitz-Stegun 7.1.26 (|err| <= 1.5e-7, fp32-level accuracy).
// 1x v_rcp_f32 + 1x v_exp_f32 + ~8 VALU, no EXEC-mask branches.
__device__ __forceinline__ float gelu_exact(float x) {
    const float z  = x * 0.70710678118654752f;
    const float az = fabsf(z);
    const float t  = __builtin_amdgcn_rcpf(fmaf(0.3275911f, az, 1.0f));
    float p = 1.061405429f;
    p = fmaf(p, t, -1.453152027f);
    p = fmaf(p, t,  1.421413741f);
    p = fmaf(p, t, -0.284496736f);
    p = fmaf(p, t,  0.254829592f);
    const float e       = __expf(-az * az);           // v_exp_f32
    const float erf_abs = fmaf(-p * t, e, 1.0f);      // saturates to 1 for large |z|
    const float erf_z   = copysignf(erf_abs, z);      // v_bfi
    return 0.5f * x * (1.0f + erf_z);
}

__global__ __launch_bounds__(TPB)
void ScoreBlock_kernel(const float* __restrict__ feats,   // [B, 1502]
                       const float* __restrict__ w1in,    // [B, 500]
                       const float* __restrict__ W1,      // [5,5]
                       const float* __restrict__ b1,      // [5]
                       const float* __restrict__ W2,      // [5,5]
                       const float* __restrict__ b2,      // [5]
                       const float* __restrict__ W3,      // [5,1]
                       const float* __restrict__ b3,      // [1]
                       const float* __restrict__ theta,   // [1]
                       float* __restrict__ out,           // [B*500 + 1]
                       int batch)
{
    const int b    = blockIdx.x;
    const int tid  = threadIdx.x;
    const int wave = tid >> 5;
    const int lane = tid & 31;
    const bool lo  = lane < 16;
    const int col  = lane & 15;

    __shared__ float ldsT[WAVES][32][17];   // padded transpose buffers (bank-conflict free)
    __shared__ float redMax[WAVES];
    __shared__ float redSum[WAVES];
    __shared__ float bcast[2];

    const float* row = feats + (size_t)b * FEAT;
    const float idx0 = row[0];
    const float idx1 = row[1];

    // ---- uniform WMMA B operands (weights) ----
    v2f B1v, B2v, B2bv;
    {
        float w1a = 0.f, w1b = 0.f, w2a = 0.f, w2b = 0.f, w2r4 = 0.f;
        if (col < 5) {
            const int r0 = lo ? 0 : 2, r1 = lo ? 1 : 3;
            w1a = W1[r0 * 5 + col]; w1b = W1[r1 * 5 + col];
            w2a = W2[r0 * 5 + col]; w2b = W2[r1 * 5 + col];
            w2r4 = lo ? W2[4 * 5 + col] : 0.0f;   // K=0 row of the fold-in WMMA
        }
        B1v.x = w1a; B1v.y = w1b;
        B2v.x = w2a; B2v.y = w2b;
        B2bv.x = w2r4; B2bv.y = 0.0f;
    }

    // ---- C operands: layer1 folds idx1*W1[4] + b1 (row-constant); layer2 C = b2 ----
    float c1v = 0.0f, c2v = 0.0f;
    if (col < 5) {
        c1v = idx1 * W1[4 * 5 + col] + b1[col];
        c2v = b2[col];
    }
    const v8f C1 = {c1v, c1v, c1v, c1v, c1v, c1v, c1v, c1v};
    const v8f C2 = {c2v, c2v, c2v, c2v, c2v, c2v, c2v, c2v};

    float w3r[5];
    #pragma unroll
    for (int k = 0; k < 5; ++k) w3r[k] = W3[k];
    const float bias3 = b3[0];
    const float omega = 1.0f / (1.0f + __expf(-theta[0]));

    float score[2];
    float ev[2];

    #pragma unroll
    for (int it = 0; it < 2; ++it) {
        const int group = wave * 2 + it;        // 0..15 -> 512 stocks covered
        const int stock = group * 32 + lane;
        const int sl    = stock < NSTOCK ? stock : (NSTOCK - 1);  // clamp OOB loads
        const float* sp = row + 2 + 3 * sl;
        const float f0 = sp[0], f1 = sp[1], f2 = sp[2];

        // ---- stage A operands across lanes (ds_bpermute) ----
        const float s2 = __shfl(f2, lane & 15, 32);         // hi lanes: f2 of lane-16
        const float s0 = __shfl(f0, (lane + 16) & 31, 32);  // lo lanes: f0 of lane+16
        const float s1 = __shfl(f1, (lane + 16) & 31, 32);
        v2f A0, A1;
        A0.x = lo ? f0 : s2;  A0.y = lo ? f1 : idx0;        // tile0: stocks group*32+0..15
        A1.x = lo ? s0 : f2;  A1.y = lo ? s1 : idx0;        // tile1: stocks group*32+16..31

        // ---- layer 1: one WMMA per 16-stock tile (K=4 fully used, 5th input in C) ----
        v8f D0 = __builtin_amdgcn_wmma_f32_16x16x4_f32(false, A0, false, B1v, (short)0, C1, false, false);
        v8f D1 = __builtin_amdgcn_wmma_f32_16x16x4_f32(false, A1, false, B1v, (short)0, C1, false, false);

        // ---- transpose D (col-per-lane) -> stock-per-lane via LDS; exact GELU x5 ----
        const int rbase = lo ? 0 : 8;
        #pragma unroll
        for (int j = 0; j < 8; ++j) {
            ldsT[wave][rbase + j][col]      = D0[j];
            ldsT[wave][16 + rbase + j][col] = D1[j];
        }
        __builtin_amdgcn_wave_barrier();
        float g[5];
        #pragma unroll
        for (int k = 0; k < 5; ++k) g[k] = gelu_exact(ldsT[wave][lane][k]);
        __builtin_amdgcn_wave_barrier();

        // ---- layer 2: two chained WMMAs per tile (K=g0..g3, then K=g4 vs W2 row4) ----
        const float t2 = __shfl(g[2], lane & 15, 32);
        const float t3 = __shfl(g[3], lane & 15, 32);
        const float t0 = __shfl(g[0], (lane + 16) & 31, 32);
        const float t1 = __shfl(g[1], (lane + 16) & 31, 32);
        const float t4 = __shfl(g[4], (lane + 16) & 31, 32);
        v2f A20, A21, A20b, A21b;
        A20.x  = lo ? g[0] : t2;   A20.y  = lo ? g[1] : t3;
        A21.x  = lo ? t0   : g[2]; A21.y  = lo ? t1   : g[3];
        A20b.x = lo ? g[4] : 0.f;  A20b.y = 0.f;
        A21b.x = lo ? t4   : 0.f;  A21b.y = 0.f;

        v8f E0 = __builtin_amdgcn_wmma_f32_16x16x4_f32(false, A20,  false, B2v,  (short)0, C2, false, false);
        E0     = __builtin_amdgcn_wmma_f32_16x16x4_f32(false, A20b, false, B2bv, (short)0, E0, false, false);
        v8f E1 = __builtin_amdgcn_wmma_f32_16x16x4_f32(false, A21,  false, B2v,  (short)0, C2, false, false);
        E1     = __builtin_amdgcn_wmma_f32_16x16x4_f32(false, A21b, false, B2bv, (short)0, E1, false, false);

        #pragma unroll
        for (int j = 0; j < 8; ++j) {
            ldsT[wave][rbase + j][col]      = E0[j];
            ldsT[wave][16 + rbase + j][col] = E1[j];
        }
        __builtin_amdgcn_wave_barrier();
        // ---- layer 3 (5x1) scalar on transposed data ----
        float sacc = bias3;
        #pragma unroll
        for (int k = 0; k < 5; ++k)
            sacc = fmaf(gelu_exact(ldsT[wave][lane][k]), w3r[k], sacc);
        __builtin_amdgcn_wave_barrier();

        score[it] = (stock < NSTOCK) ? sacc : -INFINITY;
    }

    // ---- softmax over {0} U {s_0..s_499} : block reduce max ----
    float m = fmaxf(score[0], score[1]);
    #pragma unroll
    for (int off = 16; off > 0; off >>= 1) m = fmaxf(m, __shfl_xor(m, off, 32));
    if (lane == 0) redMax[wave] = m;
    __syncthreads();
    if (tid == 0) {
        float mm = 0.0f;                      // implicit zero-score entry
        for (int w = 0; w < WAVES; ++w) mm = fmaxf(mm, redMax[w]);
        bcast[0] = mm;
    }
    __syncthreads();
    const float M = bcast[0];

    ev[0] = __expf(score[0] - M);             // -inf -> 0
    ev[1] = __expf(score[1] - M);
    float ssum = ev[0] + ev[1];
    #pragma unroll
    for (int off = 16; off > 0; off >>= 1) ssum += __shfl_xor(ssum, off, 32);
    if (lane == 0) redSum[wave] = ssum;
    __syncthreads();
    if (tid == 0) {
        float tot = __expf(-M);               // zero-score entry
        for (int w = 0; w < WAVES; ++w) tot += redSum[w];
        bcast[1] = 1.0f / tot;
    }
    __syncthreads();
    const float inv = bcast[1];

    // ---- blend with w_1 and store ----
    #pragma unroll
    for (int it = 0; it < 2; ++it) {
        const int stock = (wave * 2 + it) * 32 + lane;
        if (stock < NSTOCK) {
            const float wsc = ev[it] * inv;
            const float w2v = (1.0f - omega) * wsc
                            + omega * w1in[(size_t)b * NSTOCK + stock];
            out[(size_t)b * NSTOCK + stock] = w2v;
        }
    }
    if (b == 0 && tid == 0) out[(size_t)batch * NSTOCK] = omega;  // second tuple output
}

extern "C" void kernel_launch(void* const* d_in, const int* in_sizes, int n_in,
                              void* d_out, int out_size, void* d_ws, size_t ws_size,
                              hipStream_t stream) {
    const float* feats = (const float*)d_in[0];
    const float* w1in  = (const float*)d_in[1];
    const float* W1    = (const float*)d_in[2];
    const float* b1    = (const float*)d_in[3];
    const float* W2    = (const float*)d_in[4];
    const float* b2    = (const float*)d_in[5];
    const float* W3    = (const float*)d_in[6];
    const float* b3    = (const float*)d_in[7];
    const float* th    = (const float*)d_in[8];
    float* out = (float*)d_out;

    const int batch = in_sizes[0] / FEAT;   // 32768
    ScoreBlock_kernel<<<batch, TPB, 0, stream>>>(feats, w1in, W1, b1, W2, b2,
                                                 W3, b3, th, out, batch);
}